// PointNetSetAbstraction_17085379904242
// MI455X (gfx1250) — compile-verified
//
#include <hip/hip_runtime.h>

typedef __attribute__((ext_vector_type(2))) float v2f;
typedef __attribute__((ext_vector_type(8))) float v8f;

#define NB    8
#define NPTS  16384
#define DCH   64
#define NS    2048
#define KNN   32
#define KNNP  33           // padded list stride (bank spread)
#define BN_EPS 1e-5f

// ---------------------------------------------------------------------------
// Kernel 1: per-point squared norms  norms[b*N+n] = sum_c points[b,n,c]^2
// ---------------------------------------------------------------------------
__global__ void __launch_bounds__(256) k_norms(const float* __restrict__ points,
                                               float* __restrict__ norms) {
    int id = blockIdx.x * 256 + threadIdx.x;          // 0 .. NB*NPTS-1
    const float4* p = (const float4*)(points + (size_t)id * DCH);
    float s = 0.f;
#pragma unroll
    for (int i = 0; i < 16; i++) {
        float4 q = p[i];
        s += q.x * q.x + q.y * q.y + q.z * q.z + q.w * q.w;
    }
    norms[id] = s;
}

// ---------------------------------------------------------------------------
// Kernel 2: gather new_xyz (-> d_out) and new_pts (-> ws)
// ---------------------------------------------------------------------------
__global__ void __launch_bounds__(256) k_gather(const float* __restrict__ xyz,
                                                const float* __restrict__ points,
                                                const int* __restrict__ sidx,
                                                float* __restrict__ out_xyz,
                                                float* __restrict__ spts) {
    int id = blockIdx.x * 256 + threadIdx.x;          // 0 .. NB*NS-1
    int b = id / NS, s = id % NS;
    int si = sidx[s];
    const float* xs = xyz + ((size_t)b * NPTS + si) * 3;
    float* xd = out_xyz + (size_t)id * 3;
    xd[0] = xs[0]; xd[1] = xs[1]; xd[2] = xs[2];
    const float4* ps = (const float4*)(points + ((size_t)b * NPTS + si) * DCH);
    float4* pd = (float4*)(spts + (size_t)id * DCH);
#pragma unroll
    for (int i = 0; i < 16; i++) pd[i] = ps[i];
}

// ---------------------------------------------------------------------------
// Kernel 3: WMMA distance GEMM + top-32 selection.
// Block = 128 thr (4 waves), grid = (NS/16, NB). Sample tile (B operand) is
// pre-scaled by -2 in LDS, so WMMA accumulates -2*(s.p); selection key =
// |p_n|^2 - 2 s.p. A fragments + norms are double-buffered across tiles.
// Per-lane top-32 lists live in LDS (md/mi); only cmax/mpos in registers.
// ---------------------------------------------------------------------------
__device__ __forceinline__ void knn_issue(const float* __restrict__ pb,
                                          const float* __restrict__ nb_,
                                          int n0, int jl, int half,
                                          float2 av[16], float4& nv0, float4& nv1) {
    const float* arow = pb + (size_t)(n0 + jl) * DCH + half * 2;
#pragma unroll
    for (int k = 0; k < 16; k++) av[k] = *(const float2*)(arow + 4 * k);
    nv0 = *(const float4*)(nb_ + n0 + half * 8);
    nv1 = *(const float4*)(nb_ + n0 + half * 8 + 4);
}

__device__ __forceinline__ void topk_upd(float key, int ci,
                                         float* md, int* mi, int base,
                                         float& cmax, int& mpos) {
    if (key < cmax) {                       // rare after warm-up
        md[base + mpos] = key;
        mi[base + mpos] = ci;
        float mx = -3.4e38f; int mp = 0;
        for (int i = 0; i < KNN; i++) {
            float d = md[base + i];
            if (d > mx) { mx = d; mp = i; }
        }
        cmax = mx; mpos = mp;
    }
}

__device__ __forceinline__ void knn_process(const v2f bf[16], const float2 av[16],
                                            const float4 nv0, const float4 nv1,
                                            int n0, int half,
                                            float* md, int* mi, int base,
                                            float& cmax, int& mpos) {
    v8f c = {};
#pragma unroll
    for (int k = 0; k < 16; k++) {
        v2f a; a.x = av[k].x; a.y = av[k].y;
        c = __builtin_amdgcn_wmma_f32_16x16x4_f32(
                false, a, false, bf[k], (short)0, c, false, false);
    }
    const int cb = n0 + half * 8;           // straight-line: nothing to re-roll
    topk_upd(nv0.x + c[0], cb + 0, md, mi, base, cmax, mpos);
    topk_upd(nv0.y + c[1], cb + 1, md, mi, base, cmax, mpos);
    topk_upd(nv0.z + c[2], cb + 2, md, mi, base, cmax, mpos);
    topk_upd(nv0.w + c[3], cb + 3, md, mi, base, cmax, mpos);
    topk_upd(nv1.x + c[4], cb + 4, md, mi, base, cmax, mpos);
    topk_upd(nv1.y + c[5], cb + 5, md, mi, base, cmax, mpos);
    topk_upd(nv1.z + c[6], cb + 6, md, mi, base, cmax, mpos);
    topk_upd(nv1.w + c[7], cb + 7, md, mi, base, cmax, mpos);
}

__global__ void __launch_bounds__(128) k_knn(const float* __restrict__ points,
                                             const float* __restrict__ norms,
                                             const float* __restrict__ spts,
                                             int* __restrict__ gidx) {
    __shared__ float sp[16][68];             // sample tile * (-2), padded
    __shared__ float md[8 * 16 * KNNP];      // per-lane top-32 dist lists
    __shared__ int   mi[8 * 16 * KNNP];      // per-lane top-32 index lists
    __shared__ float smin[16][8];
    __shared__ int   sminp[16][8];

    const int b   = blockIdx.y;
    const int s0  = blockIdx.x * 16;
    const int tid = threadIdx.x;
    const int lane = tid & 31, wave = tid >> 5;

    {   // stage 16x64 sample-feature tile to LDS, scaled by -2
        int r = tid >> 3, c0 = (tid & 7) * 8;
        const float* src = spts + ((size_t)b * NS + s0 + r) * DCH + c0;
#pragma unroll
        for (int u = 0; u < 8; u++) sp[r][c0 + u] = -2.f * src[u];
    }
    __syncthreads();

    const int half = lane >> 4, jl = lane & 15;

    // loop-invariant B fragments (sample columns) -> registers
    v2f bf[16];
    {
        const float* brow = &sp[jl][half * 2];
#pragma unroll
        for (int k = 0; k < 16; k++) {
            float2 bv = *(const float2*)(brow + 4 * k);
            bf[k].x = bv.x; bf[k].y = bv.y;
        }
    }

    // init this lane's private LDS list region
    const int base = ((wave * 2 + half) * 16 + jl) * KNNP;
#pragma unroll
    for (int i = 0; i < KNN; i++) { md[base + i] = 3.4e38f; mi[base + i] = 0; }
    float cmax = 3.4e38f; int mpos = 0;

    const int chunk = NPTS / 4;
    const int nbeg = wave * chunk, nend = nbeg + chunk;
    const float* nb_ = norms + (size_t)b * NPTS;
    const float* pb  = points + (size_t)b * NPTS * DCH;

    float2 avA[16], avB[16];
    float4 nA0, nA1, nB0, nB1;
    knn_issue(pb, nb_, nbeg, jl, half, avA, nA0, nA1);
    for (int n0 = nbeg; n0 < nend; n0 += 32) {
        knn_issue(pb, nb_, n0 + 16, jl, half, avB, nB0, nB1);
        knn_process(bf, avA, nA0, nA1, n0, half, md, mi, base, cmax, mpos);
        int np = (n0 + 32 < nend) ? (n0 + 32) : nbeg;   // safe dummy prefetch
        knn_issue(pb, nb_, np, jl, half, avA, nA0, nA1);
        knn_process(bf, avB, nB0, nB1, n0 + 16, half, md, mi, base, cmax, mpos);
    }
    __syncthreads();

    // parallel merge of 8 partial top-32 lists per sample -> final 32.
    // Order irrelevant downstream (max-pool is permutation invariant).
    const int j = tid >> 3;       // sample 0..15
    const int t = tid & 7;        // group  0..7
    for (int r = 0; r < KNN; r++) {
        {
            int off = (t * 16 + j) * KNNP;
            float best = 3.5e38f; int bp = off;
            for (int i = 0; i < KNN; i++) {
                float d = md[off + i];
                if (d < best) { best = d; bp = off + i; }
            }
            smin[j][t] = best; sminp[j][t] = bp;
        }
        __syncthreads();
        if (tid < 16) {
            float best = 3.5e38f; int bp = 0;
#pragma unroll
            for (int t2 = 0; t2 < 8; t2++) {
                float d = smin[tid][t2];
                if (d < best) { best = d; bp = sminp[tid][t2]; }
            }
            gidx[((size_t)b * NS + s0 + tid) * KNN + r] = mi[bp];
            md[bp] = 3.5e38f;
        }
        __syncthreads();
    }
}

// ---------------------------------------------------------------------------
// Kernel 4: gather 32x64 neighbor features, 3 WMMA MLP layers (+BN+ReLU),
// max-pool over neighbors. Block = 256 thr (8 waves) per (b,s).
// ---------------------------------------------------------------------------
__device__ __forceinline__ void mlp_layer(const float* in, int instr,
                                          float* out, int outstr,
                                          const float* __restrict__ W,
                                          const float* __restrict__ bb,
                                          const float* __restrict__ gg,
                                          const float* __restrict__ bt,
                                          const float* __restrict__ mm,
                                          const float* __restrict__ vv,
                                          int O, int lane, int wave) {
    const int half = lane >> 4, jl = lane & 15;
    const int ntiles = 2 * (O / 16);
    for (int t = wave; t < ntiles; t += 8) {
        const int mt = t & 1, ot = t >> 1;
        const float* Ab = in + (mt * 16 + jl) * instr + half * 2;
        const float* Bb = W + (size_t)(ot * 16 + jl) * DCH + half * 2;
        const int o = ot * 16 + jl;
        // prefetch weight fragments + BN params (batch the global loads)
        float2 wv[16];
#pragma unroll
        for (int k = 0; k < 16; k++) wv[k] = *(const float2*)(Bb + 4 * k);
        const float sc = gg[o] * rsqrtf(vv[o] + BN_EPS);
        const float sh = fmaf(bb[o] - mm[o], sc, bt[o]);
        v8f c = {};
#pragma unroll
        for (int k = 0; k < 16; k++) {
            float2 av = *(const float2*)(Ab + 4 * k);
            v2f a;  a.x  = av.x;    a.y  = av.y;
            v2f bq; bq.x = wv[k].x; bq.y = wv[k].y;
            c = __builtin_amdgcn_wmma_f32_16x16x4_f32(
                    false, a, false, bq, (short)0, c, false, false);
        }
        float* orow = out + (size_t)mt * 16 * outstr + half * 8 * outstr + o;
#define STORE_ROW(V) { float y = fmaf(c[V], sc, sh); \
                       orow[(V) * outstr] = y > 0.f ? y : 0.f; }
        STORE_ROW(0) STORE_ROW(1) STORE_ROW(2) STORE_ROW(3)
        STORE_ROW(4) STORE_ROW(5) STORE_ROW(6) STORE_ROW(7)
#undef STORE_ROW
    }
}

__global__ void __launch_bounds__(256) k_mlp(const float* __restrict__ points,
                                             const int* __restrict__ gidx,
        const float* w0, const float* b0, const float* g0, const float* bt0,
        const float* mu0, const float* va0,
        const float* w1, const float* b1, const float* g1, const float* bt1,
        const float* mu1, const float* va1,
        const float* w2, const float* b2, const float* g2, const float* bt2,
        const float* mu2, const float* va2,
        float* __restrict__ outp) {
    __shared__ float bufA[32][68];    // 64-channel activations
    __shared__ float bufB[32][136];   // up-to-128-channel activations
    const int b = blockIdx.y, s = blockIdx.x;
    const int tid = threadIdx.x, lane = tid & 31, wave = tid >> 5;

    {   // gather grouped 32x64 neighbor features into LDS
        int r = tid >> 3, c0 = (tid & 7) * 8;
        int ii = gidx[((size_t)b * NS + s) * KNN + r];
        const float* src = points + ((size_t)b * NPTS + ii) * DCH + c0;
#pragma unroll
        for (int u = 0; u < 8; u++) bufA[r][c0 + u] = src[u];
    }
    __syncthreads();

    mlp_layer(&bufA[0][0], 68, &bufB[0][0], 136, w0, b0, g0, bt0, mu0, va0, 64, lane, wave);
    __syncthreads();
    mlp_layer(&bufB[0][0], 136, &bufA[0][0], 68, w1, b1, g1, bt1, mu1, va1, 64, lane, wave);
    __syncthreads();
    mlp_layer(&bufA[0][0], 68, &bufB[0][0], 136, w2, b2, g2, bt2, mu2, va2, 128, lane, wave);
    __syncthreads();

    if (tid < 128) {   // max-pool over 32 neighbors
        float mx = -3.4e38f;
#pragma unroll 4
        for (int r = 0; r < 32; r++) mx = fmaxf(mx, bufB[r][tid]);
        outp[((size_t)b * NS + s) * 128 + tid] = mx;
    }
}

// ---------------------------------------------------------------------------
extern "C" void kernel_launch(void* const* d_in, const int* in_sizes, int n_in,
                              void* d_out, int out_size, void* d_ws, size_t ws_size,
                              hipStream_t stream) {
    (void)in_sizes; (void)n_in; (void)out_size; (void)ws_size;
    const float* xyz    = (const float*)d_in[0];
    const float* points = (const float*)d_in[1];
    const int*   sidx   = (const int*)d_in[2];
    const float* w0  = (const float*)d_in[3];
    const float* b0  = (const float*)d_in[4];
    const float* g0  = (const float*)d_in[5];
    const float* bt0 = (const float*)d_in[6];
    const float* mu0 = (const float*)d_in[7];
    const float* va0 = (const float*)d_in[8];
    const float* w1  = (const float*)d_in[9];
    const float* b1  = (const float*)d_in[10];
    const float* g1  = (const float*)d_in[11];
    const float* bt1 = (const float*)d_in[12];
    const float* mu1 = (const float*)d_in[13];
    const float* va1 = (const float*)d_in[14];
    const float* w2  = (const float*)d_in[15];
    const float* b2  = (const float*)d_in[16];
    const float* g2  = (const float*)d_in[17];
    const float* bt2 = (const float*)d_in[18];
    const float* mu2 = (const float*)d_in[19];
    const float* va2 = (const float*)d_in[20];

    float* out = (float*)d_out;
    char*  ws  = (char*)d_ws;
    float* norms = (float*)ws;                                  // 512 KB
    float* spts  = (float*)(ws + 524288);                       // 4 MB
    int*   gidx  = (int*)(ws + 524288 + 4194304);               // 2 MB
    float* out_xyz = out;
    float* out_pts = out + (size_t)NB * NS * 3;

    k_norms <<<NB * NPTS / 256, 256, 0, stream>>>(points, norms);
    k_gather<<<NB * NS   / 256, 256, 0, stream>>>(xyz, points, sidx, out_xyz, spts);
    k_knn   <<<dim3(NS / 16, NB), 128, 0, stream>>>(points, norms, spts, gidx);
    k_mlp   <<<dim3(NS, NB), 256, 0, stream>>>(points, gidx,
            w0, b0, g0, bt0, mu0, va0,
            w1, b1, g1, bt1, mu1, va1,
            w2, b2, g2, bt2, mu2, va2,
            out_pts);
}